// OurSE3Transformer_90726889161256
// MI455X (gfx1250) — compile-verified
//
#include <hip/hip_runtime.h>
#include <math.h>

// ---------------------------------------------------------------------------
// SE3-Transformer forward for MI455X (gfx1250).
//  * All 123 radial GEMMs ([E,8]x[8,32] padded, [E,32]x[32,32],
//    [E,32]x[32,co*ci*F]) run through v_wmma_f32_16x16x4_f32 with
//    compile-time K (fully unrolled, no EXEC masking, clause'd loads).
//  * Basis contractions / attention are streaming VALU kernels with float
//    atomics for segment max / segment sum.
//  * Workspace: bump allocator over d_ws (~650 MB needed).
// ---------------------------------------------------------------------------

typedef __attribute__((ext_vector_type(2))) float v2f;
typedef __attribute__((ext_vector_type(8))) float v8f;

#define DEV __device__ __forceinline__

static __host__ __device__ inline int mdeg(int d) { return 2 * d + 1; }

// ------------------------------ GEMM (WMMA) --------------------------------
// D[M,N] = A[M,4*KT] @ B[4*KT,N] + bias[N].  N multiple of 16, M multiple of
// 64.  4 waves/block, each wave owns a 16-row strip and sweeps all N tiles;
// A fragments hoisted out of the N loop.  No divergence: EXEC stays all-1s
// (WMMA requirement), K loop fully unrolled, bias unconditional.
template <int KT>
__global__ void wmma_gemm_bias(const float* __restrict__ A, int lda,
                               const float* __restrict__ B, int ldb,
                               const float* __restrict__ bias,
                               float* __restrict__ D, int ldd, int N) {
  const int lane = threadIdx.x & 31;
  const int wave = threadIdx.x >> 5;
  const int row0 = blockIdx.x * 64 + wave * 16;
  const int l15 = lane & 15;
  const int half = lane >> 4;  // 0: lanes 0-15, 1: lanes 16-31
  const int arow = row0 + l15;

  // A fragment: ISA 32-bit A 16x4 layout. VGPR0=K(2*half), VGPR1=K(2*half+1).
  v2f afrag[KT];
  {
    const float* ar = A + (size_t)arow * lda + 2 * half;
#pragma unroll
    for (int k4 = 0; k4 < KT; ++k4) {
      afrag[k4].x = ar[k4 * 4 + 0];
      afrag[k4].y = ar[k4 * 4 + 1];
    }
  }

  const int ntiles = N >> 4;
  for (int nt = 0; nt < ntiles; ++nt) {
    const int col = nt * 16 + l15;
    const float* bc = B + (size_t)(2 * half) * ldb + col;
    const float bv = bias[col];
    v8f c = {0.f, 0.f, 0.f, 0.f, 0.f, 0.f, 0.f, 0.f};
#pragma unroll
    for (int k4 = 0; k4 < KT; ++k4) {
      v2f bfrag;
      bfrag.x = bc[(size_t)(k4 * 4) * ldb];
      bfrag.y = bc[(size_t)(k4 * 4 + 1) * ldb];
      c = __builtin_amdgcn_wmma_f32_16x16x4_f32(false, afrag[k4], false, bfrag,
                                                (short)0, c, false, false);
    }
#pragma unroll
    for (int r = 0; r < 8; ++r) {
      const int row = row0 + r + 8 * half;  // C/D layout: row = r + 8*(l/16)
      D[(size_t)row * ldd + col] = c[r] + bv;
    }
  }
}

// Pad per-pair w1 [5,32] into w1p [8,32] (rows 5..7 zero).  One block.
__global__ void pad_w1_kernel(const float* __restrict__ w1,
                              float* __restrict__ w1p) {
  const int t = threadIdx.x;  // 256 threads = 8*32
  const int r = t >> 5, c = t & 31;
  w1p[t] = (r < 5) ? w1[r * 32 + c] : 0.f;
}

// -------------------------- LayerNorm(32) + ReLU ---------------------------
__global__ void ln_relu32(float* __restrict__ X, const float* __restrict__ g,
                          const float* __restrict__ be, int M) {
  const int lane = threadIdx.x & 31;
  const int row = blockIdx.x * (blockDim.x >> 5) + (threadIdx.x >> 5);
  if (row >= M) return;
  float v = X[(size_t)row * 32 + lane];
  float s = v;
#pragma unroll
  for (int m = 16; m >= 1; m >>= 1) s += __shfl_xor(s, m, 32);
  const float mu = s * (1.f / 32.f);
  const float dv = v - mu;
  float q = dv * dv;
#pragma unroll
  for (int m = 16; m >= 1; m >>= 1) q += __shfl_xor(q, m, 32);
  const float var = q * (1.f / 32.f);
  float o = dv * rsqrtf(var + 1e-5f) * g[lane] + be[lane];
  X[(size_t)row * 32 + lane] = fmaxf(o, 0.f);
}

// ------------------------- Edge message contraction ------------------------
// msg[e,c,a] += sum_{i,f} R[e,c,i,f] * (sum_b basis[e,a,b,f] * h[src[e],i,b])
template <int C, int I, int F, int MI, int MO>
__global__ void edge_msg(const float* __restrict__ R,
                         const float* __restrict__ hsrc,
                         const int* __restrict__ src,
                         const float* __restrict__ basis,
                         float* __restrict__ msg, int E) {
  const int e = blockIdx.x * blockDim.x + threadIdx.x;
  if (e >= E) return;
  const float* hrow = hsrc + (size_t)src[e] * (I * MI);
  const float* Re = R + (size_t)e * (C * I * F);
  const float* Be = basis + (size_t)e * (MO * MI * F);
  float acc[C * MO];
#pragma unroll
  for (int t = 0; t < C * MO; ++t) acc[t] = 0.f;
#pragma unroll
  for (int f = 0; f < F; ++f) {
    float bas[MO * MI];
#pragma unroll
    for (int a = 0; a < MO; ++a)
#pragma unroll
      for (int b = 0; b < MI; ++b) bas[a * MI + b] = Be[(a * MI + b) * F + f];
    for (int i = 0; i < I; ++i) {
      float ft[MI];
#pragma unroll
      for (int b = 0; b < MI; ++b) ft[b] = hrow[i * MI + b];
      float tv[MO];
#pragma unroll
      for (int a = 0; a < MO; ++a) {
        float s = 0.f;
#pragma unroll
        for (int b = 0; b < MI; ++b) s += bas[a * MI + b] * ft[b];
        tv[a] = s;
      }
#pragma unroll
      for (int c = 0; c < C; ++c) {
        const float rv = Re[(c * I + i) * F + f];
#pragma unroll
        for (int a = 0; a < MO; ++a) acc[c * MO + a] += rv * tv[a];
      }
    }
  }
  float* mo_ = msg + (size_t)e * (C * MO);
#pragma unroll
  for (int t = 0; t < C * MO; ++t) mo_[t] += acc[t];
}

// ------------------------------ small kernels ------------------------------
// Writes rfeat padded to 8 columns: [w0..w3, r, 0, 0, 0].
__global__ void rfeat_kernel(const float* __restrict__ x,
                             const float* __restrict__ w,
                             const int* __restrict__ src,
                             const int* __restrict__ dst,
                             float* __restrict__ rfeat, int E) {
  const int e = blockIdx.x * blockDim.x + threadIdx.x;
  if (e >= E) return;
  const float* xs = x + 3 * (size_t)src[e];
  const float* xd = x + 3 * (size_t)dst[e];
  const float dx = xd[0] - xs[0], dy = xd[1] - xs[1], dz = xd[2] - xs[2];
  const float r = sqrtf(dx * dx + dy * dy + dz * dz + 1e-12f);
  float* o = rfeat + (size_t)e * 8;
  o[0] = w[(size_t)e * 4 + 0];
  o[1] = w[(size_t)e * 4 + 1];
  o[2] = w[(size_t)e * 4 + 2];
  o[3] = w[(size_t)e * 4 + 3];
  o[4] = r;
  o[5] = 0.f;
  o[6] = 0.f;
  o[7] = 0.f;
}

__global__ void qproj_kernel(const float* __restrict__ W,  // [4, I]
                             const float* __restrict__ h,  // [N, I*m]
                             float* __restrict__ q,        // [N, 4*m]
                             int I, int m, int Nn) {
  const int t = blockIdx.x * blockDim.x + threadIdx.x;
  const int tot = Nn * 4 * m;
  if (t >= tot) return;
  const int a = t % m, c = (t / m) % 4, n = t / (4 * m);
  const float* hr = h + (size_t)n * I * m;
  float s = 0.f;
  for (int i = 0; i < I; ++i) s += W[c * I + i] * hr[i * m + a];
  q[t] = s;
}

struct Deg4 {
  const float* k[4];
  const float* q[4];
  int m[4];
};

__global__ void logits_kernel(Deg4 dd, const int* __restrict__ dst,
                              float inv_sqrt_dk, float* __restrict__ logits,
                              int E) {
  const int e = blockIdx.x * blockDim.x + threadIdx.x;
  if (e >= E) return;
  const int dn = dst[e];
  float l0 = 0.f, l1 = 0.f;
  for (int d = 0; d < 4; ++d) {
    const int m = dd.m[d];
    if (m == 0) continue;
    const float* kr = dd.k[d] + (size_t)e * 4 * m;
    const float* qr = dd.q[d] + (size_t)dn * 4 * m;
    for (int c = 0; c < 4; ++c) {
      float s = 0.f;
      for (int a = 0; a < m; ++a) s += kr[c * m + a] * qr[c * m + a];
      if (c < 2) l0 += s; else l1 += s;
    }
  }
  logits[e * 2 + 0] = l0 * inv_sqrt_dk;
  logits[e * 2 + 1] = l1 * inv_sqrt_dk;
}

DEV unsigned enc_f(float f) {
  const unsigned u = __float_as_uint(f);
  return (u & 0x80000000u) ? ~u : (u | 0x80000000u);
}
DEV float dec_f(unsigned u) {
  return (u & 0x80000000u) ? __uint_as_float(u & 0x7fffffffu)
                           : __uint_as_float(~u);
}

__global__ void segmax_kernel(const float* __restrict__ logits,
                              const int* __restrict__ dst,
                              unsigned* __restrict__ mxu, int E) {
  const int t = blockIdx.x * blockDim.x + threadIdx.x;
  if (t >= E * 2) return;
  const int e = t >> 1, hh = t & 1;
  atomicMax(&mxu[(size_t)dst[e] * 2 + hh], enc_f(logits[t]));
}

__global__ void mxfin_kernel(const unsigned* __restrict__ mxu,
                             float* __restrict__ mxf, int n2) {
  const int t = blockIdx.x * blockDim.x + threadIdx.x;
  if (t >= n2) return;
  float m = dec_f(mxu[t]);
  if (!(m > -3.0e38f && m < 3.0e38f)) m = 0.f;  // NaN/inf -> 0 (ref semantics)
  mxf[t] = m;
}

__global__ void expsum_kernel(const float* __restrict__ logits,
                              const float* __restrict__ mxf,
                              const int* __restrict__ dst,
                              float* __restrict__ aexp,
                              float* __restrict__ den, int E) {
  const int t = blockIdx.x * blockDim.x + threadIdx.x;
  if (t >= E * 2) return;
  const int e = t >> 1, hh = t & 1;
  const float a = expf(logits[t] - mxf[(size_t)dst[e] * 2 + hh]);
  aexp[t] = a;
  atomicAdd(&den[(size_t)dst[e] * 2 + hh], a);
}

__global__ void alpha_kernel(float* __restrict__ aexp,
                             const float* __restrict__ den,
                             const int* __restrict__ dst, int E) {
  const int t = blockIdx.x * blockDim.x + threadIdx.x;
  if (t >= E * 2) return;
  const int e = t >> 1, hh = t & 1;
  aexp[t] = aexp[t] / fmaxf(den[(size_t)dst[e] * 2 + hh], 1e-9f);
}

__global__ void scatter_v(const float* __restrict__ v,
                          const float* __restrict__ alpha,
                          const int* __restrict__ dst, float* __restrict__ z,
                          int m, int E) {
  const int e = blockIdx.x * blockDim.x + threadIdx.x;
  if (e >= E) return;
  const int dn = dst[e];
  const float* vr = v + (size_t)e * 4 * m;
  float* zr = z + (size_t)dn * 4 * m;
  const float a0 = alpha[e * 2], a1 = alpha[e * 2 + 1];
  for (int c = 0; c < 4; ++c) {
    const float al = (c < 2) ? a0 : a1;
    for (int a = 0; a < m; ++a) atomicAdd(&zr[c * m + a], al * vr[c * m + a]);
  }
}

__global__ void proj_kernel(const float* __restrict__ z,     // [N,4*m]
                            const float* __restrict__ hold,  // [N,hc*m]|null
                            const float* __restrict__ W,     // [16, 4+hc]
                            float* __restrict__ hnew,        // [N,16*m]
                            int m, int hc, int Nn) {
  const int t = blockIdx.x * blockDim.x + threadIdx.x;
  const int tot = Nn * 16 * m;
  if (t >= tot) return;
  const int a = t % m, o = (t / m) % 16, n = t / (16 * m);
  const float* wr = W + o * (4 + hc);
  const float* zr = z + (size_t)n * 4 * m;
  float s = 0.f;
  for (int c = 0; c < 4; ++c) s += wr[c] * zr[c * m + a];
  if (hold) {
    const float* hr = hold + (size_t)n * hc * m;
    for (int c = 0; c < hc; ++c) s += wr[4 + c] * hr[c * m + a];
  }
  hnew[t] = s;
}

__global__ void gnorm_kernel(float* __restrict__ h, const float* __restrict__ g,
                             const float* __restrict__ be,
                             const float* __restrict__ W,
                             const float* __restrict__ b, int m, int Nn) {
  const int n = blockIdx.x * blockDim.x + threadIdx.x;
  if (n >= Nn) return;
  float* hr = h + (size_t)n * 16 * m;
  float nrm[16], t[16], t2[16];
#pragma unroll
  for (int c = 0; c < 16; ++c) {
    float s = 1e-12f;
    for (int a = 0; a < m; ++a) {
      const float v = hr[c * m + a];
      s += v * v;
    }
    nrm[c] = sqrtf(s);
  }
  float mu = 0.f;
#pragma unroll
  for (int c = 0; c < 16; ++c) mu += nrm[c];
  mu *= (1.f / 16.f);
  float var = 0.f;
#pragma unroll
  for (int c = 0; c < 16; ++c) {
    const float d = nrm[c] - mu;
    var += d * d;
  }
  var *= (1.f / 16.f);
  const float inv = rsqrtf(var + 1e-5f);
#pragma unroll
  for (int c = 0; c < 16; ++c)
    t[c] = fmaxf((nrm[c] - mu) * inv * g[c] + be[c], 0.f);
#pragma unroll
  for (int o = 0; o < 16; ++o) {
    float s = b[o];
#pragma unroll
    for (int c = 0; c < 16; ++c) s += t[c] * W[c * 16 + o];
    t2[o] = s;
  }
#pragma unroll
  for (int c = 0; c < 16; ++c) {
    const float sc = t2[c] / nrm[c];
    for (int a = 0; a < m; ++a) hr[c * m + a] *= sc;
  }
}

__global__ void cnt_kernel(const int* __restrict__ dst, float* __restrict__ cnt,
                           int E) {
  const int e = blockIdx.x * blockDim.x + threadIdx.x;
  if (e >= E) return;
  atomicAdd(&cnt[dst[e]], 1.f);
}

__global__ void mean_scatter(const float* __restrict__ msgF,
                             const int* __restrict__ dst,
                             float* __restrict__ meanF, int E) {
  const int t = blockIdx.x * blockDim.x + threadIdx.x;
  if (t >= E * 64) return;
  const int c = t & 63, e = t >> 6;
  atomicAdd(&meanF[(size_t)dst[e] * 64 + c], msgF[t]);
}

__global__ void out0_kernel(const float* __restrict__ meanF,
                            const float* __restrict__ cnt,
                            const float* __restrict__ h00,  // [N,16]
                            const float* __restrict__ w_self,
                            const float* __restrict__ st, float* __restrict__ z0,
                            int Nn) {
  const int t = blockIdx.x * blockDim.x + threadIdx.x;
  if (t >= Nn * 64) return;
  const int c = t & 63, n = t >> 6;
  const float cn = cnt[n];
  const float mv = meanF[t] / fmaxf(cn, 1.f);
  float wh = 0.f;
  const float* hr = h00 + (size_t)n * 16;
#pragma unroll
  for (int i = 0; i < 16; ++i) wh += w_self[c * 16 + i] * hr[i];
  const float o = mv + ((cn > 0.f) ? wh : 0.f);
  z0[t] = fmaxf(o * st[0], 0.f);
}

__global__ void fc_kernel(const float* __restrict__ in,
                          const float* __restrict__ W,
                          const float* __restrict__ b, float* __restrict__ out,
                          int M, int I, int O, int relu) {
  const int t = blockIdx.x * blockDim.x + threadIdx.x;
  if (t >= M * O) return;
  const int o = t % O, n = t / O;
  const float* ir = in + (size_t)n * I;
  float s = b[o];
  for (int i = 0; i < I; ++i) s += ir[i] * W[i * O + o];
  out[t] = relu ? fmaxf(s, 0.f) : s;
}

// ------------------------------- host side ---------------------------------
extern "C" void kernel_launch(void* const* d_in, const int* in_sizes, int n_in,
                              void* d_out, int out_size, void* d_ws,
                              size_t ws_size, hipStream_t stream) {
  (void)n_in; (void)out_size; (void)ws_size;
  const int E = in_sizes[3];        // src length
  const int Nn = in_sizes[0] / 32;  // h0 = [N,32,1]

  // ---- walk d_in in recursive dict-insertion order ----
  int idx = 0;
  auto FP = [&]() { return (const float*)d_in[idx++]; };
  const float* h0 = FP();
  const float* x = FP();
  const float* wE = FP();
  const int* src = (const int*)d_in[idx++];
  const int* dst = (const int*)d_in[idx++];
  const float* basis[4][4];
  for (int di = 0; di < 4; ++di)
    for (int dp = 0; dp < 4; ++dp) basis[di][dp] = FP();
  const float* scalar_trick = FP();

  struct Radial {
    const float *w1, *b1, *g1, *be1, *w2, *b2, *g2, *be2, *w3, *b3;
  };
  auto RD = [&]() {
    Radial r;
    r.w1 = FP(); r.b1 = FP(); r.g1 = FP(); r.be1 = FP();
    r.w2 = FP(); r.b2 = FP(); r.g2 = FP(); r.be2 = FP();
    r.w3 = FP(); r.b3 = FP();
    return r;
  };
  struct GN { const float *g, *be, *w, *b; };
  auto GRD = [&]() { GN g; g.g = FP(); g.be = FP(); g.w = FP(); g.b = FP(); return g; };

  Radial r0v[4];
  for (int dp = 0; dp < 4; ++dp) r0v[dp] = RD();   // res0.v: '0,do'
  Radial r0k = RD();                               // res0.k: '0,0'
  const float* r0q = FP();                         // res0.q['0']
  const float* r0proj[4];
  for (int d = 0; d < 4; ++d) r0proj[d] = FP();
  GN n0[4];
  for (int d = 0; d < 4; ++d) n0[d] = GRD();
  Radial r1v[4][4];                                // [do][di], do-outer order
  for (int dp = 0; dp < 4; ++dp)
    for (int di = 0; di < 4; ++di) r1v[dp][di] = RD();
  Radial r1k[4][4];
  for (int dp = 0; dp < 4; ++dp)
    for (int di = 0; di < 4; ++di) r1k[dp][di] = RD();
  const float* r1q[4];
  for (int d = 0; d < 4; ++d) r1q[d] = FP();
  const float* r1proj[4];
  for (int d = 0; d < 4; ++d) r1proj[d] = FP();
  GN n1[4];
  for (int d = 0; d < 4; ++d) n1[d] = GRD();
  Radial fin[4];
  for (int di = 0; di < 4; ++di) fin[di] = RD();
  const float* w_self = FP();
  const float *fcw[4], *fcb[4];
  for (int i = 0; i < 4; ++i) { fcw[i] = FP(); fcb[i] = FP(); }

  // ---- workspace bump allocator ----
  char* wp = (char*)d_ws;
  auto alloc = [&](size_t nfl) {
    float* p = (float*)wp;
    wp += ((nfl * 4 + 255) / 256) * 256;
    return p;
  };
  float* rfeat = alloc((size_t)E * 8);  // padded to K=8
  float* w1p = alloc(8 * 32);           // padded per-pair w1
  float* h1 = alloc((size_t)E * 32);
  float* h2 = alloc((size_t)E * 32);
  float* Rbuf = alloc((size_t)E * 1024);
  float *msgv[4], *msgk[4], *qb[4], *zb[4], *hA[4], *hB[4];
  for (int d = 0; d < 4; ++d) {
    const int m = mdeg(d);
    msgv[d] = alloc((size_t)E * 4 * m);
    msgk[d] = alloc((size_t)E * 4 * m);
    qb[d] = alloc((size_t)Nn * 4 * m);
    zb[d] = alloc((size_t)Nn * 4 * m);
    hA[d] = alloc((size_t)Nn * 16 * m);
    hB[d] = alloc((size_t)Nn * 16 * m);
  }
  float* logits = alloc((size_t)E * 2);
  float* aexp = alloc((size_t)E * 2);
  unsigned* mxu = (unsigned*)alloc((size_t)Nn * 2);
  float* mxf = alloc((size_t)Nn * 2);
  float* den = alloc((size_t)Nn * 2);
  float* msgF = alloc((size_t)E * 64);
  float* meanF = alloc((size_t)Nn * 64);
  float* cnt = alloc((size_t)Nn);
  float* z0 = alloc((size_t)Nn * 64);
  float* z1 = alloc((size_t)Nn * 32);
  float* z2 = alloc((size_t)Nn * 16);
  float* z3 = alloc((size_t)Nn * 8);

  const dim3 B256(256);
  const dim3 gE((E + 255) / 256);
  auto cfg = [&](int n) { return dim3((n + 255) / 256); };
  auto zero = [&](void* p, size_t bytes) { hipMemsetAsync(p, 0, bytes, stream); };

  auto gemm8 = [&](const float* A, int lda, const float* Bm, int ldb,
                   const float* bias, float* D, int ldd, int M, int N) {
    wmma_gemm_bias<2><<<dim3(M / 64), dim3(128), 0, stream>>>(A, lda, Bm, ldb,
                                                              bias, D, ldd, N);
  };
  auto gemm32 = [&](const float* A, int lda, const float* Bm, int ldb,
                    const float* bias, float* D, int ldd, int M, int N) {
    wmma_gemm_bias<8><<<dim3(M / 64), dim3(128), 0, stream>>>(A, lda, Bm, ldb,
                                                              bias, D, ldd, N);
  };
  auto lnrelu = [&](float* X, const float* g, const float* be, int M) {
    ln_relu32<<<dim3((M + 7) / 8), B256, 0, stream>>>(X, g, be, M);
  };
  auto radial = [&](const Radial& p, int cif, float* Rout) {
    pad_w1_kernel<<<dim3(1), B256, 0, stream>>>(p.w1, w1p);
    gemm8(rfeat, 8, w1p, 32, p.b1, h1, 32, E, 32);
    lnrelu(h1, p.g1, p.be1, E);
    gemm32(h1, 32, p.w2, 32, p.b2, h2, 32, E, 32);
    lnrelu(h2, p.g2, p.be2, E);
    gemm32(h2, 32, p.w3, cif, p.b3, Rout, cif, E, cif);
  };

  auto msg_l0 = [&](int mo, const float* Rp, const float* hs, const float* bas,
                    float* mg) {
    switch (mo) {
      case 1: edge_msg<4, 32, 1, 1, 1><<<gE, B256, 0, stream>>>(Rp, hs, src, bas, mg, E); break;
      case 3: edge_msg<4, 32, 1, 1, 3><<<gE, B256, 0, stream>>>(Rp, hs, src, bas, mg, E); break;
      case 5: edge_msg<4, 32, 1, 1, 5><<<gE, B256, 0, stream>>>(Rp, hs, src, bas, mg, E); break;
      default: edge_msg<4, 32, 1, 1, 7><<<gE, B256, 0, stream>>>(Rp, hs, src, bas, mg, E); break;
    }
  };
  auto msg_l1 = [&](int di, int dp, const float* Rp, const float* hs,
                    const float* bas, float* mg) {
    switch (di * 4 + dp) {
      case 0:  edge_msg<4, 16, 1, 1, 1><<<gE, B256, 0, stream>>>(Rp, hs, src, bas, mg, E); break;
      case 1:  edge_msg<4, 16, 1, 1, 3><<<gE, B256, 0, stream>>>(Rp, hs, src, bas, mg, E); break;
      case 2:  edge_msg<4, 16, 1, 1, 5><<<gE, B256, 0, stream>>>(Rp, hs, src, bas, mg, E); break;
      case 3:  edge_msg<4, 16, 1, 1, 7><<<gE, B256, 0, stream>>>(Rp, hs, src, bas, mg, E); break;
      case 4:  edge_msg<4, 16, 1, 3, 1><<<gE, B256, 0, stream>>>(Rp, hs, src, bas, mg, E); break;
      case 5:  edge_msg<4, 16, 3, 3, 3><<<gE, B256, 0, stream>>>(Rp, hs, src, bas, mg, E); break;
      case 6:  edge_msg<4, 16, 3, 3, 5><<<gE, B256, 0, stream>>>(Rp, hs, src, bas, mg, E); break;
      case 7:  edge_msg<4, 16, 3, 3, 7><<<gE, B256, 0, stream>>>(Rp, hs, src, bas, mg, E); break;
      case 8:  edge_msg<4, 16, 1, 5, 1><<<gE, B256, 0, stream>>>(Rp, hs, src, bas, mg, E); break;
      case 9:  edge_msg<4, 16, 3, 5, 3><<<gE, B256, 0, stream>>>(Rp, hs, src, bas, mg, E); break;
      case 10: edge_msg<4, 16, 5, 5, 5><<<gE, B256, 0, stream>>>(Rp, hs, src, bas, mg, E); break;
      case 11: edge_msg<4, 16, 5, 5, 7><<<gE, B256, 0, stream>>>(Rp, hs, src, bas, mg, E); break;
      case 12: edge_msg<4, 16, 1, 7, 1><<<gE, B256, 0, stream>>>(Rp, hs, src, bas, mg, E); break;
      case 13: edge_msg<4, 16, 3, 7, 3><<<gE, B256, 0, stream>>>(Rp, hs, src, bas, mg, E); break;
      case 14: edge_msg<4, 16, 5, 7, 5><<<gE, B256, 0, stream>>>(Rp, hs, src, bas, mg, E); break;
      default: edge_msg<4, 16, 7, 7, 7><<<gE, B256, 0, stream>>>(Rp, hs, src, bas, mg, E); break;
    }
  };
  auto msg_fin = [&](int di, const float* Rp, const float* hs, const float* bas,
                     float* mg) {
    switch (di) {
      case 0: edge_msg<64, 16, 1, 1, 1><<<gE, B256, 0, stream>>>(Rp, hs, src, bas, mg, E); break;
      case 1: edge_msg<64, 16, 1, 3, 1><<<gE, B256, 0, stream>>>(Rp, hs, src, bas, mg, E); break;
      case 2: edge_msg<64, 16, 1, 5, 1><<<gE, B256, 0, stream>>>(Rp, hs, src, bas, mg, E); break;
      default: edge_msg<64, 16, 1, 7, 1><<<gE, B256, 0, stream>>>(Rp, hs, src, bas, mg, E); break;
    }
  };

  auto attention = [&](Deg4 dd, float inv_sqrt_dk) {
    logits_kernel<<<gE, B256, 0, stream>>>(dd, dst, inv_sqrt_dk, logits, E);
    zero(mxu, (size_t)Nn * 2 * 4);
    segmax_kernel<<<cfg(E * 2), B256, 0, stream>>>(logits, dst, mxu, E);
    mxfin_kernel<<<cfg(Nn * 2), B256, 0, stream>>>(mxu, mxf, Nn * 2);
    zero(den, (size_t)Nn * 2 * 4);
    expsum_kernel<<<cfg(E * 2), B256, 0, stream>>>(logits, mxf, dst, aexp, den, E);
    alpha_kernel<<<cfg(E * 2), B256, 0, stream>>>(aexp, den, dst, E);
    for (int d = 0; d < 4; ++d) {
      const int m = mdeg(d);
      zero(zb[d], (size_t)Nn * 4 * m * 4);
      scatter_v<<<gE, B256, 0, stream>>>(msgv[d], aexp, dst, zb[d], m, E);
    }
  };

  // ============================ forward pass ===============================
  rfeat_kernel<<<gE, B256, 0, stream>>>(x, wE, src, dst, rfeat, E);

  // ---- layer 0 (in_degs = {0}, in_ch = 32) ----
  for (int dp = 0; dp < 4; ++dp) {
    zero(msgv[dp], (size_t)E * 4 * mdeg(dp) * 4);
    radial(r0v[dp], 4 * 32 * 1, Rbuf);
    msg_l0(mdeg(dp), Rbuf, h0, basis[0][dp], msgv[dp]);
  }
  zero(msgk[0], (size_t)E * 4 * 4);
  radial(r0k, 4 * 32 * 1, Rbuf);
  msg_l0(1, Rbuf, h0, basis[0][0], msgk[0]);
  qproj_kernel<<<cfg(Nn * 4), B256, 0, stream>>>(r0q, h0, qb[0], 32, 1, Nn);
  {
    Deg4 dd = {};
    dd.k[0] = msgk[0]; dd.q[0] = qb[0]; dd.m[0] = 1;
    attention(dd, 0.5f);  // dk = 4
  }
  for (int d = 0; d < 4; ++d) {
    const int m = mdeg(d);
    proj_kernel<<<cfg(Nn * 16 * m), B256, 0, stream>>>(
        zb[d], d == 0 ? h0 : (const float*)nullptr, r0proj[d], hA[d], m,
        d == 0 ? 32 : 0, Nn);
    gnorm_kernel<<<cfg(Nn), B256, 0, stream>>>(hA[d], n0[d].g, n0[d].be,
                                               n0[d].w, n0[d].b, m, Nn);
  }

  // ---- layer 1 (in_degs = {0..3}, in_ch = 16) ----
  for (int dp = 0; dp < 4; ++dp) {
    zero(msgv[dp], (size_t)E * 4 * mdeg(dp) * 4);
    for (int di = 0; di < 4; ++di) {
      const int F = 2 * (di < dp ? di : dp) + 1;
      radial(r1v[dp][di], 4 * 16 * F, Rbuf);
      msg_l1(di, dp, Rbuf, hA[di], basis[di][dp], msgv[dp]);
    }
  }
  for (int dp = 0; dp < 4; ++dp) {
    zero(msgk[dp], (size_t)E * 4 * mdeg(dp) * 4);
    for (int di = 0; di < 4; ++di) {
      const int F = 2 * (di < dp ? di : dp) + 1;
      radial(r1k[dp][di], 4 * 16 * F, Rbuf);
      msg_l1(di, dp, Rbuf, hA[di], basis[di][dp], msgk[dp]);
    }
  }
  for (int d = 0; d < 4; ++d) {
    const int m = mdeg(d);
    qproj_kernel<<<cfg(Nn * 4 * m), B256, 0, stream>>>(r1q[d], hA[d], qb[d],
                                                       16, m, Nn);
  }
  {
    Deg4 dd = {};
    for (int d = 0; d < 4; ++d) { dd.k[d] = msgk[d]; dd.q[d] = qb[d]; dd.m[d] = mdeg(d); }
    attention(dd, 0.125f);  // dk = 64
  }
  for (int d = 0; d < 4; ++d) {
    const int m = mdeg(d);
    proj_kernel<<<cfg(Nn * 16 * m), B256, 0, stream>>>(zb[d], hA[d], r1proj[d],
                                                       hB[d], m, 16, Nn);
    gnorm_kernel<<<cfg(Nn), B256, 0, stream>>>(hB[d], n1[d].g, n1[d].be,
                                               n1[d].w, n1[d].b, m, Nn);
  }

  // ---- final conv (degree-0 out, mean aggregation) + head ----
  zero(msgF, (size_t)E * 64 * 4);
  for (int di = 0; di < 4; ++di) {
    radial(fin[di], 64 * 16 * 1, Rbuf);
    msg_fin(di, Rbuf, hB[di], basis[di][0], msgF);
  }
  zero(cnt, (size_t)Nn * 4);
  cnt_kernel<<<gE, B256, 0, stream>>>(dst, cnt, E);
  zero(meanF, (size_t)Nn * 64 * 4);
  mean_scatter<<<cfg(E * 64), B256, 0, stream>>>(msgF, dst, meanF, E);
  out0_kernel<<<cfg(Nn * 64), B256, 0, stream>>>(meanF, cnt, hB[0], w_self,
                                                 scalar_trick, z0, Nn);
  fc_kernel<<<cfg(Nn * 32), B256, 0, stream>>>(z0, fcw[0], fcb[0], z1, Nn, 64, 32, 1);
  fc_kernel<<<cfg(Nn * 16), B256, 0, stream>>>(z1, fcw[1], fcb[1], z2, Nn, 32, 16, 1);
  fc_kernel<<<cfg(Nn * 8), B256, 0, stream>>>(z2, fcw[2], fcb[2], z3, Nn, 16, 8, 1);
  fc_kernel<<<cfg(Nn * 3), B256, 0, stream>>>(z3, fcw[3], fcb[3], (float*)d_out,
                                              Nn, 8, 3, 0);
}